// GTLayer_29180007809572
// MI455X (gfx1250) — compile-verified
//
#include <hip/hip_runtime.h>

typedef __attribute__((ext_vector_type(2))) float v2f;
typedef __attribute__((ext_vector_type(8))) float v8f;

#define DIM 64
#define HEADS 4
#define DH 16

// ---------------------------------------------------------------------------
// Zero-fill helper (res accumulator and softmax denominators must start at 0)
// ---------------------------------------------------------------------------
__global__ void zero_f32(float* __restrict__ p, int n) {
    int i = blockIdx.x * blockDim.x + threadIdx.x;
    if (i < n) p[i] = 0.0f;
}

// ---------------------------------------------------------------------------
// Node projection: Q = E@Wq, K = E@Wk, V = E@Wv   ([N,64] x [64,64])
// One wave handles a 16-row x 16-col output tile for all three matrices.
// K-dim 64 = 16 steps of v_wmma_f32_16x16x4_f32.
//
// Per ISA 7.12.2 (32-bit operands, wave32):
//   A (16x4, MxK): lanes 0-15 -> M=lane, VGPR0=K0, VGPR1=K1
//                  lanes 16-31 -> M=lane-16, VGPR0=K2, VGPR1=K3
//   B (4x16, KxN): lanes 0-15 -> N=lane, VGPR0=K0, VGPR1=K1
//                  lanes 16-31 -> N=lane-16, VGPR0=K2, VGPR1=K3
//   C/D (16x16):   VGPR r: lanes 0-15 -> M=r, N=lane
//                          lanes 16-31 -> M=r+8, N=lane-16
// ---------------------------------------------------------------------------
__global__ void __launch_bounds__(128)
proj_wmma(const float* __restrict__ embeds,
          const float* __restrict__ Wq,
          const float* __restrict__ Wk,
          const float* __restrict__ Wv,
          float* __restrict__ Q,
          float* __restrict__ K,
          float* __restrict__ V,
          int nNodes)
{
    const int lane = threadIdx.x & 31;
    const int wave = threadIdx.x >> 5;     // 4 waves -> 4 column tiles of 16
    const int m0   = blockIdx.x * 16;      // node tile base row
    const int n0   = wave * 16;            // output column tile base
    if (m0 >= nNodes) return;

    const int mA   = lane & 15;                 // A row / B,C column within tile
    const int kSel = (lane < 16) ? 0 : 2;       // K pair selected by lane half
    const int rowA = min(m0 + mA, nNodes - 1);  // clamped gather row (guard)

    v8f cq = {};
    v8f ck = {};
    v8f cv = {};

#pragma unroll
    for (int kk = 0; kk < 16; ++kk) {
        const int kb = kk * 4 + kSel;

        v2f a;
        a.x = embeds[rowA * DIM + kb];
        a.y = embeds[rowA * DIM + kb + 1];

        v2f bq, bk, bv;
        bq.x = Wq[kb * DIM + n0 + mA];
        bq.y = Wq[(kb + 1) * DIM + n0 + mA];
        bk.x = Wk[kb * DIM + n0 + mA];
        bk.y = Wk[(kb + 1) * DIM + n0 + mA];
        bv.x = Wv[kb * DIM + n0 + mA];
        bv.y = Wv[(kb + 1) * DIM + n0 + mA];

        cq = __builtin_amdgcn_wmma_f32_16x16x4_f32(false, a, false, bq, (short)0, cq, false, false);
        ck = __builtin_amdgcn_wmma_f32_16x16x4_f32(false, a, false, bk, (short)0, ck, false, false);
        cv = __builtin_amdgcn_wmma_f32_16x16x4_f32(false, a, false, bv, (short)0, cv, false, false);
    }

#pragma unroll
    for (int r = 0; r < 8; ++r) {
        const int row = m0 + ((lane < 16) ? r : r + 8);
        const int col = n0 + mA;
        if (row < nNodes) {
            Q[row * DIM + col] = cq[r];
            K[row * DIM + col] = ck[r];
            V[row * DIM + col] = cv[r];
        }
    }
}

// ---------------------------------------------------------------------------
// Per (edge, head): att logit = <q[row], k[col]> over 16 dims, clip, exp.
// Store expAtt into attOut; atomicAdd denominator per (row, head).
// ---------------------------------------------------------------------------
__global__ void __launch_bounds__(256)
edge_logits(const float* __restrict__ Q,
            const float* __restrict__ K,
            const int* __restrict__ rows,
            const int* __restrict__ cols,
            float* __restrict__ attOut,   // [E, H]  (holds expAtt after this)
            float* __restrict__ norm,     // [N, H]
            int E)
{
    int t = blockIdx.x * blockDim.x + threadIdx.x;
    if (t >= E * HEADS) return;
    const int e = t >> 2;
    const int h = t & 3;

    const int r = rows[e];
    const int c = cols[e];

    const float4* qp = (const float4*)(Q + (size_t)r * DIM + h * DH);
    const float4* kp = (const float4*)(K + (size_t)c * DIM + h * DH);

    float acc = 0.0f;
#pragma unroll
    for (int i = 0; i < 4; ++i) {
        float4 qa = qp[i];
        float4 kb = kp[i];
        acc += qa.x * kb.x + qa.y * kb.y + qa.z * kb.z + qa.w * kb.w;
    }
    acc = fminf(10.0f, fmaxf(-10.0f, acc));
    float ex = expf(acc);

    attOut[(size_t)e * HEADS + h] = ex;
    atomicAdd(&norm[(size_t)r * HEADS + h], ex);
}

// ---------------------------------------------------------------------------
// Per (edge, head): att = expAtt / (norm[row] + 1e-8); write final att;
// scatter att * v[col] into res[row] with f32 atomics.
// ---------------------------------------------------------------------------
__global__ void __launch_bounds__(256)
norm_scatter(const float* __restrict__ V,
             const int* __restrict__ rows,
             const int* __restrict__ cols,
             const float* __restrict__ norm,
             float* __restrict__ attOut,   // in: expAtt, out: normalized att
             float* __restrict__ res,      // [N, DIM] accumulator
             int E)
{
    int t = blockIdx.x * blockDim.x + threadIdx.x;
    if (t >= E * HEADS) return;
    const int e = t >> 2;
    const int h = t & 3;

    const int r = rows[e];
    const int c = cols[e];

    const float a = attOut[(size_t)e * HEADS + h] /
                    (norm[(size_t)r * HEADS + h] + 1e-8f);
    attOut[(size_t)e * HEADS + h] = a;

    const float4* vp = (const float4*)(V + (size_t)c * DIM + h * DH);
    float* rp = res + (size_t)r * DIM + h * DH;

#pragma unroll
    for (int i = 0; i < 4; ++i) {
        float4 vv = vp[i];
        atomicAdd(rp + 4 * i + 0, a * vv.x);
        atomicAdd(rp + 4 * i + 1, a * vv.y);
        atomicAdd(rp + 4 * i + 2, a * vv.z);
        atomicAdd(rp + 4 * i + 3, a * vv.w);
    }
}

// ---------------------------------------------------------------------------
// Launch
// ---------------------------------------------------------------------------
extern "C" void kernel_launch(void* const* d_in, const int* in_sizes, int n_in,
                              void* d_out, int out_size, void* d_ws, size_t ws_size,
                              hipStream_t stream) {
    const float* embeds = (const float*)d_in[0];
    const float* Wq     = (const float*)d_in[1];
    const float* Wk     = (const float*)d_in[2];
    const float* Wv     = (const float*)d_in[3];
    const int*   rows   = (const int*)d_in[4];
    const int*   cols   = (const int*)d_in[5];

    const int nNodes = in_sizes[0] / DIM;
    const int E      = in_sizes[4];

    // Workspace layout: Q | K | V | norm
    float* Q    = (float*)d_ws;
    float* K    = Q + (size_t)nNodes * DIM;
    float* V    = K + (size_t)nNodes * DIM;
    float* norm = V + (size_t)nNodes * DIM;

    // Output layout (tuple return order): res [N, DIM] then att [E, HEADS]
    float* res = (float*)d_out;
    float* att = res + (size_t)nNodes * DIM;

    // 1) zero accumulators
    {
        int n = nNodes * DIM;
        zero_f32<<<(n + 255) / 256, 256, 0, stream>>>(res, n);
        n = nNodes * HEADS;
        zero_f32<<<(n + 255) / 256, 256, 0, stream>>>(norm, n);
    }

    // 2) node projections via WMMA (f32 16x16x4)
    {
        int nTiles = (nNodes + 15) / 16;
        proj_wmma<<<nTiles, 128, 0, stream>>>(embeds, Wq, Wk, Wv, Q, K, V, nNodes);
    }

    // 3) edge logits -> expAtt + denominators
    {
        int n = E * HEADS;
        edge_logits<<<(n + 255) / 256, 256, 0, stream>>>(Q, K, rows, cols, att, norm, E);
    }

    // 4) normalize + weighted scatter into res
    {
        int n = E * HEADS;
        norm_scatter<<<(n + 255) / 256, 256, 0, stream>>>(V, rows, cols, norm, att, res, E);
    }
}